// GraphClassifier_44624710205923
// MI455X (gfx1250) — compile-verified
//
#include <hip/hip_runtime.h>

typedef __attribute__((ext_vector_type(2))) float v2f;
typedef __attribute__((ext_vector_type(8))) float v8f;

namespace {
constexpr int  cB   = 512;
constexpr int  cN   = 200;
constexpr int  cDEG = 32;
constexpr int  cDIM = 64;
constexpr int  cK1  = 160;
constexpr int  cK2  = 128;
constexpr long cE   = (long)cB * cN * cDEG;   // 3,276,800
constexpr int  cBN  = cB * cN;                // 102,400
constexpr int  cBK1 = cB * cK1;               // 81,920
constexpr int  cBK2 = cB * cK2;               // 65,536
}

// ---------------------------------------------------------------------------
// fp32 WMMA GEMM: one wave32 per 16x16 output tile, K stepped by 4.
// A: MxK row-major, B: KxN row-major, C: MxN row-major.
// A layout (16x4 f32): lanes 0-15 -> M=lane, K={k,k+1}; lanes 16-31 -> K={k+2,k+3}
// B layout (4x16 f32): VGPR0/1 = rows k+koff / k+koff+1 striped on lanes 0-15 (N),
//                      lanes 16-31 carry rows k+2/k+3.
// D layout: VGPR r -> M = r (lanes 0-15) or r+8 (lanes 16-31), N = lane&15.
// ---------------------------------------------------------------------------
__global__ void k_gemm(const float* __restrict__ A, const float* __restrict__ Bm,
                       float* __restrict__ C, int M, int N, int K) {
  int wid  = (blockIdx.x * blockDim.x + threadIdx.x) >> 5;
  int lane = threadIdx.x & 31;
  int ntiles = N >> 4;
  int mt = wid / ntiles;
  int nt = wid - mt * ntiles;
  if (mt >= (M >> 4)) return;                      // wave-uniform: EXEC stays all-ones
  const int rowA = mt * 16 + (lane & 15);
  const int col  = nt * 16 + (lane & 15);
  const int koff = (lane >> 4) * 2;
  v8f acc = {};
  for (int k = 0; k < K; k += 4) {
    v2f a, b;
    const float* ap = A + (long)rowA * K + (k + koff);
    a.x = ap[0];
    a.y = ap[1];
    const float* bp = Bm + (long)(k + koff) * N + col;
    b.x = bp[0];
    b.y = bp[N];
    acc = __builtin_amdgcn_wmma_f32_16x16x4_f32(false, a, false, b, (short)0, acc,
                                                false, false);
  }
  const int r0 = mt * 16 + ((lane >> 4) * 8);
#pragma unroll
  for (int r = 0; r < 8; ++r) C[(long)(r0 + r) * N + col] = acc[r];
}

// GEMM with fused  bn(relu(A@B + bias))  epilogue (eval-mode BatchNorm1d).
__global__ void k_gemm_fc(const float* __restrict__ A, const float* __restrict__ Bm,
                          float* __restrict__ C, int M, int N, int K,
                          const float* __restrict__ bias,
                          const float* __restrict__ bng, const float* __restrict__ bnb,
                          const float* __restrict__ bnm, const float* __restrict__ bnv) {
  int wid  = (blockIdx.x * blockDim.x + threadIdx.x) >> 5;
  int lane = threadIdx.x & 31;
  int ntiles = N >> 4;
  int mt = wid / ntiles;
  int nt = wid - mt * ntiles;
  if (mt >= (M >> 4)) return;
  const int rowA = mt * 16 + (lane & 15);
  const int col  = nt * 16 + (lane & 15);
  const int koff = (lane >> 4) * 2;
  v8f acc = {};
  for (int k = 0; k < K; k += 4) {
    v2f a, b;
    const float* ap = A + (long)rowA * K + (k + koff);
    a.x = ap[0];
    a.y = ap[1];
    const float* bp = Bm + (long)(k + koff) * N + col;
    b.x = bp[0];
    b.y = bp[N];
    acc = __builtin_amdgcn_wmma_f32_16x16x4_f32(false, a, false, b, (short)0, acc,
                                                false, false);
  }
  const float bi = bias[col];
  const float sc = bng[col] * rsqrtf(bnv[col] + 1e-5f);
  const float mm = bnm[col];
  const float ob = bnb[col];
  const int r0 = mt * 16 + ((lane >> 4) * 8);
#pragma unroll
  for (int r = 0; r < 8; ++r) {
    float v = fmaxf(acc[r] + bi, 0.f);
    C[(long)(r0 + r) * N + col] = (v - mm) * sc + ob;
  }
}

// ---------------------------------------------------------------------------
// 1/||p|| for a 64-vector (one wave)
__global__ void k_inv_norm(const float* __restrict__ p, float* __restrict__ out, int n) {
  int lane = threadIdx.x;
  float acc = 0.f;
  for (int i = lane; i < n; i += 32) { float v = p[i]; acc += v * v; }
  for (int off = 16; off; off >>= 1) acc += __shfl_down(acc, off, 32);
  if (lane == 0) out[0] = rsqrtf(acc);
}

// conv1 scatter: agg[dst] += xw[src]  (16 threads/edge, float4 each)
__global__ void k_scatter1(const int* __restrict__ src, const int* __restrict__ dst,
                           const float* __restrict__ xw, float* __restrict__ agg,
                           long nE) {
  long tid = (long)blockIdx.x * blockDim.x + threadIdx.x;
  long e = tid >> 4;
  if (e >= nE) return;
  int c = (int)(tid & 15) * 4;
  int s = src[e], d = dst[e];
  const float4 m = *(const float4*)(xw + (long)s * cDIM + c);
  float* o = agg + (long)d * cDIM + c;
  atomicAdd(o + 0, m.x); atomicAdd(o + 1, m.y);
  atomicAdd(o + 2, m.z); atomicAdd(o + 3, m.w);
}

// conv2 scatter with node remap + validity filter
__global__ void k_scatter2(const int* __restrict__ src, const int* __restrict__ dst,
                           const int* __restrict__ noo, const float* __restrict__ hw,
                           float* __restrict__ agg, long nE) {
  long tid = (long)blockIdx.x * blockDim.x + threadIdx.x;
  long e = tid >> 4;
  if (e >= nE) return;
  int s = noo[src[e]], d = noo[dst[e]];
  if (s < 0 || d < 0) return;
  int c = (int)(tid & 15) * 4;
  const float4 m = *(const float4*)(hw + (long)s * cDIM + c);
  float* o = agg + (long)d * cDIM + c;
  atomicAdd(o + 0, m.x); atomicAdd(o + 1, m.y);
  atomicAdd(o + 2, m.z); atomicAdd(o + 3, m.w);
}

// per node: h = relu(agg + bias) in place; score = tanh(dot(h,p)/||p||)  (wave/node)
__global__ void k_finish_node(float* __restrict__ h, const float* __restrict__ bias,
                              const float* __restrict__ p, const float* __restrict__ invn,
                              float* __restrict__ score, int nNodes) {
  int wid  = (blockIdx.x * blockDim.x + threadIdx.x) >> 5;
  int lane = threadIdx.x & 31;
  if (wid >= nNodes) return;
  float* row = h + (long)wid * cDIM;
  float dotv = 0.f;
  for (int i = lane; i < cDIM; i += 32) {
    float v = fmaxf(row[i] + bias[i], 0.f);
    row[i] = v;
    dotv += v * p[i];
  }
  for (int off = 16; off; off >>= 1) dotv += __shfl_down(dotv, off, 32);
  if (lane == 0) score[wid] = tanhf(dotv * invn[0]);
}

// exact top-k by rank (matches lax.top_k ordering, stable on ties). block/graph.
__global__ void k_pool(const float* __restrict__ score, int n, int k,
                       float* __restrict__ vals, int* __restrict__ oon,
                       int* __restrict__ noo, float* __restrict__ attn) {
  int b = blockIdx.x;
  __shared__ float s[256];
  int t = threadIdx.x;
  if (t < n) s[t] = score[(long)b * n + t];
  __syncthreads();
  if (t >= n) return;
  float v = s[t];
  int rank = 0;
  for (int j = 0; j < n; ++j) {
    float u = s[j];
    rank += (u > v) || (u == v && j < t);
  }
  if (rank < k) {
    int nid = b * k + rank;
    vals[nid] = v;
    oon[nid]  = b * n + t;
    if (noo) noo[b * n + t] = nid;
    attn[nid] = 1.f / (1.f + __expf(-v));
  }
}

// xp[j] = h[old_of_new[j]] * vals[j]   (16 threads/node, float4)
__global__ void k_gather_scale(const float* __restrict__ h, const int* __restrict__ oon,
                               const float* __restrict__ vals, float* __restrict__ xp,
                               int nNew) {
  int tid = blockIdx.x * blockDim.x + threadIdx.x;
  int j = tid >> 4;
  if (j >= nNew) return;
  int c = (tid & 15) * 4;
  int o = oon[j];
  float v = vals[j];
  float4 d = *(const float4*)(h + (long)o * cDIM + c);
  d.x *= v; d.y *= v; d.z *= v; d.w *= v;
  *(float4*)(xp + (long)j * cDIM + c) = d;
}

// per graph: xcat[b][colOff + c] = max_k xp, xcat[b][colOff+64+c] = mean_k xp
__global__ void k_maxmean(const float* __restrict__ xp, int kk,
                          float* __restrict__ xcat, int colOff) {
  int b = blockIdx.x;
  int c = threadIdx.x;                 // 64 threads
  const float* base = xp + (long)b * kk * cDIM + c;
  float mx = -__builtin_inff(), sm = 0.f;
  for (int i = 0; i < kk; ++i) {
    float v = base[(long)i * cDIM];
    mx = fmaxf(mx, v);
    sm += v;
  }
  xcat[(long)b * 256 + colOff + c]      = mx;
  xcat[(long)b * 256 + colOff + 64 + c] = sm / (float)kk;
}

// fc3 (1024 -> 2) + log_softmax, one wave per row
__global__ void k_fc3(const float* __restrict__ y, const float* __restrict__ W,
                      const float* __restrict__ bias, float* __restrict__ out) {
  int row = blockIdx.x;
  int lane = threadIdx.x;
  const float* yr = y + (long)row * 1024;
  float a0 = 0.f, a1 = 0.f;
  for (int i = lane; i < 1024; i += 32) {
    float v = yr[i];
    a0 += v * W[i * 2 + 0];
    a1 += v * W[i * 2 + 1];
  }
  for (int off = 16; off; off >>= 1) {
    a0 += __shfl_down(a0, off, 32);
    a1 += __shfl_down(a1, off, 32);
  }
  if (lane == 0) {
    a0 += bias[0]; a1 += bias[1];
    float m = fmaxf(a0, a1);
    float lse = m + logf(expf(a0 - m) + expf(a1 - m));
    out[row * 2 + 0] = a0 - lse;
    out[row * 2 + 1] = a1 - lse;
  }
}

// ---------------------------------------------------------------------------
extern "C" void kernel_launch(void* const* d_in, const int* in_sizes, int n_in,
                              void* d_out, int out_size, void* d_ws, size_t ws_size,
                              hipStream_t stream) {
  (void)in_sizes; (void)n_in; (void)out_size; (void)ws_size;
  const float* x    = (const float*)d_in[0];
  const int*   eidx = (const int*)d_in[1];
  // d_in[2] = edge_attr: unused by the reference (edge_weight=None)
  const float* w1   = (const float*)d_in[3];
  const float* b1   = (const float*)d_in[4];
  const float* p1   = (const float*)d_in[5];
  const float* w2   = (const float*)d_in[6];
  const float* b2   = (const float*)d_in[7];
  const float* p2   = (const float*)d_in[8];
  const float* fc1w = (const float*)d_in[9];
  const float* fc1b = (const float*)d_in[10];
  const float* bn1g = (const float*)d_in[11];
  const float* bn1b = (const float*)d_in[12];
  const float* bn1m = (const float*)d_in[13];
  const float* bn1v = (const float*)d_in[14];
  const float* fc2w = (const float*)d_in[15];
  const float* fc2b = (const float*)d_in[16];
  const float* bn2g = (const float*)d_in[17];
  const float* bn2b = (const float*)d_in[18];
  const float* bn2m = (const float*)d_in[19];
  const float* bn2v = (const float*)d_in[20];
  const float* fc3w = (const float*)d_in[21];
  const float* fc3b = (const float*)d_in[22];

  const int* src = eidx;
  const int* dst = eidx + cE;

  // ---- workspace carve-up (256B aligned) ----
  char* ws = (char*)d_ws;
  auto carve = [&](size_t bytes) -> char* {
    char* q = ws;
    ws += (bytes + 255) & ~(size_t)255;
    return q;
  };
  float* xw1    = (float*)carve((size_t)cBN  * cDIM * 4);
  float* agg1   = (float*)carve((size_t)cBN  * cDIM * 4);   // becomes h1 in place
  float* score1 = (float*)carve((size_t)cBN  * 4);
  float* invn   = (float*)carve(256);                       // [0]=1/||p1||, [1]=1/||p2||
  float* vals1  = (float*)carve((size_t)cBK1 * 4);
  int*   oon1   = (int*)  carve((size_t)cBK1 * 4);
  int*   noo    = (int*)  carve((size_t)cBN  * 4);
  float* xp1    = (float*)carve((size_t)cBK1 * cDIM * 4);
  float* hw2    = (float*)carve((size_t)cBK1 * cDIM * 4);
  float* agg2   = (float*)carve((size_t)cBK1 * cDIM * 4);   // becomes h2 in place
  float* score2 = (float*)carve((size_t)cBK1 * 4);
  float* vals2  = (float*)carve((size_t)cBK2 * 4);
  int*   oon2   = (int*)  carve((size_t)cBK2 * 4);
  float* xp2    = (float*)carve((size_t)cBK2 * cDIM * 4);
  float* xcat   = (float*)carve((size_t)cB * 256 * 4);
  float* y1     = (float*)carve((size_t)cB * 512 * 4);
  float* y2     = (float*)carve((size_t)cB * 1024 * 4);

  float* out_xy    = (float*)d_out;                 // [512,2]
  float* out_attn1 = out_xy + cB * 2;               // [512*160]
  float* out_attn2 = out_attn1 + cBK1;              // [512*128]

  hipMemsetAsync(agg1, 0,    (size_t)cBN  * cDIM * 4, stream);
  hipMemsetAsync(agg2, 0,    (size_t)cBK1 * cDIM * 4, stream);
  hipMemsetAsync(noo,  0xFF, (size_t)cBN  * 4,        stream);   // -1

  k_inv_norm<<<1, 32, 0, stream>>>(p1, invn + 0, cDIM);
  k_inv_norm<<<1, 32, 0, stream>>>(p2, invn + 1, cDIM);

  const int edgeThreads = (int)(cE * 16);
  const int edgeBlocks  = (edgeThreads + 255) / 256;

  // ---- layer 1: GCN + ReLU + score + top-k pool + readout ----
  { int tiles = (cBN / 16) * (cDIM / 16);           // 25,600 waves
    k_gemm<<<(tiles + 7) / 8, 256, 0, stream>>>(x, w1, xw1, cBN, cDIM, 200); }
  k_scatter1<<<edgeBlocks, 256, 0, stream>>>(src, dst, xw1, agg1, cE);
  k_finish_node<<<(cBN * 32 + 255) / 256, 256, 0, stream>>>(agg1, b1, p1, invn + 0,
                                                            score1, cBN);
  k_pool<<<cB, 256, 0, stream>>>(score1, cN, cK1, vals1, oon1, noo, out_attn1);
  k_gather_scale<<<(cBK1 * 16 + 255) / 256, 256, 0, stream>>>(agg1, oon1, vals1, xp1, cBK1);
  k_maxmean<<<cB, 64, 0, stream>>>(xp1, cK1, xcat, 0);

  // ---- layer 2: GCN on pooled graph + pool + readout ----
  { int tiles = (cBK1 / 16) * (cDIM / 16);          // 20,480 waves
    k_gemm<<<(tiles + 7) / 8, 256, 0, stream>>>(xp1, w2, hw2, cBK1, cDIM, cDIM); }
  k_scatter2<<<edgeBlocks, 256, 0, stream>>>(src, dst, noo, hw2, agg2, cE);
  k_finish_node<<<(cBK1 * 32 + 255) / 256, 256, 0, stream>>>(agg2, b2, p2, invn + 1,
                                                             score2, cBK1);
  k_pool<<<cB, 256, 0, stream>>>(score2, cK1, cK2, vals2, oon2, nullptr, out_attn2);
  k_gather_scale<<<(cBK2 * 16 + 255) / 256, 256, 0, stream>>>(agg2, oon2, vals2, xp2, cBK2);
  k_maxmean<<<cB, 64, 0, stream>>>(xp2, cK2, xcat, 128);

  // ---- MLP head: fc1+relu+bn, fc2+relu+bn (fused WMMA epilogues), fc3+log_softmax
  { int tiles = (cB / 16) * (512 / 16);             // 1,024 waves
    k_gemm_fc<<<(tiles + 7) / 8, 256, 0, stream>>>(xcat, fc1w, y1, cB, 512, 256,
                                                   fc1b, bn1g, bn1b, bn1m, bn1v); }
  { int tiles = (cB / 16) * (1024 / 16);            // 2,048 waves
    k_gemm_fc<<<(tiles + 7) / 8, 256, 0, stream>>>(y1, fc2w, y2, cB, 1024, 512,
                                                   fc2b, bn2g, bn2b, bn2m, bn2v); }
  k_fc3<<<cB, 32, 0, stream>>>(y2, fc3w, fc3b, out_xy);
}